// RefineLayer_24326694765305
// MI455X (gfx1250) — compile-verified
//
#include <hip/hip_runtime.h>
#include <math.h>

// ---------------------------------------------------------------------------
// Problem constants (B,M,T,H,W) = (4,4,16,256,256), N_STEP = 5, fftconst = 256
// ---------------------------------------------------------------------------
#define BB   4
#define MM   4
#define TT   16
#define NN   256           // H == W == 256
#define HW   (NN*NN)       // 65536
#define NIMG (BB*MM*TT)    // 256 images per 5D tensor [B,M,T,H,W]
#define NOBJ ((size_t)BB*TT*HW)     // 4,194,304 elements [B,T,H,W]
#define NPE  ((size_t)NIMG*HW)      // 16,777,216 elements [B,M,T,H,W]

typedef __attribute__((ext_vector_type(2))) float v2f;
typedef __attribute__((ext_vector_type(8))) float v8f;

// D = A(16x4) * B(4x16) + C(16x16), fp32 WMMA.  NEG[0]/NEG[1] must be 0 for
// F32 per ISA (only CNeg allowed), so all sign handling is done in VALU.
#define WMMA_F32(a, b, c) \
  __builtin_amdgcn_wmma_f32_16x16x4_f32(false, (a), false, (b), (short)0, (c), false, false)

// ---------------------------------------------------------------------------
// Build shifted+scaled DFT matrices, complex-interleaved row-major.
//  Fs[k][n] = exp(-2*pi*i*(k+128)*n/256)/16   (fftshift folded, /16 per axis)
//  Gs[n][j] = exp(+2*pi*i*n*(j+128)/256)/16   (ifftshift + *fc folded)
// fwd:  dif   = Fs * X * Fs^T        (== fftshift(fft2(X))/256)
// inv:  exitw = Gs * Y * Gs^T        (== ifft2(ifftshift(Y))*256)
// ---------------------------------------------------------------------------
__global__ void init_dft_k(float* __restrict__ FsC,  float* __restrict__ FsTC,
                           float* __restrict__ GsC,  float* __restrict__ GsTC) {
  const int r = blockIdx.x;      // row
  const int c = threadIdx.x;     // col
  const float s = 1.0f / 16.0f;
  const float w = 6.283185307179586476925f / 256.0f;
  int tf = ((r + 128) * c) & 255;          // phase index mod 256 (exact)
  float fr =  cosf(w * (float)tf) * s;
  float fi = -sinf(w * (float)tf) * s;
  FsC[(r * NN + c) * 2]      = fr;  FsC[(r * NN + c) * 2 + 1]  = fi;
  FsTC[(c * NN + r) * 2]     = fr;  FsTC[(c * NN + r) * 2 + 1] = fi;
  int tg = (r * (c + 128)) & 255;
  float gr = cosf(w * (float)tg) * s;
  float gi = sinf(w * (float)tg) * s;
  GsC[(r * NN + c) * 2]      = gr;  GsC[(r * NN + c) * 2 + 1]  = gi;
  GsTC[(c * NN + r) * 2]     = gr;  GsTC[(c * NN + r) * 2 + 1] = gi;
}

// ---------------------------------------------------------------------------
// prob_abs = sum_m max_{h,w} |probe[m]|^2   (single workgroup reduction)
// ---------------------------------------------------------------------------
__global__ __launch_bounds__(256) void prob_abs_k(const float* __restrict__ pre,
                                                  const float* __restrict__ pim,
                                                  float* __restrict__ pa) {
  __shared__ float sm[256];
  const int tid = threadIdx.x;
  float total = 0.0f;
  for (int m = 0; m < MM; ++m) {
    const float* pr = pre + (size_t)m * HW;
    const float* pi = pim + (size_t)m * HW;
    float mx = 0.0f;
    for (int i = tid; i < HW; i += 256) {
      float v = pr[i] * pr[i] + pi[i] * pi[i];
      mx = fmaxf(mx, v);
    }
    sm[tid] = mx;
    __syncthreads();
    for (int st = 128; st > 0; st >>= 1) {
      if (tid < st) sm[tid] = fmaxf(sm[tid], sm[tid + st]);
      __syncthreads();
    }
    if (tid == 0) total += sm[0];
    __syncthreads();
  }
  if (tid == 0) pa[0] = total;
}

// Copy separate re/im object planes into an interleaved working buffer.
__global__ __launch_bounds__(256) void copy_obj_k(const float* __restrict__ ore,
                                                  const float* __restrict__ oim,
                                                  float* __restrict__ objW) {
  size_t i = (size_t)blockIdx.x * 256 + threadIdx.x;
  objW[2 * i]     = ore[i];
  objW[2 * i + 1] = oim[i];
}

// pre_exit[b,m,t] = probe[m] * objects[b,t]   (complex), interleaved output
__global__ __launch_bounds__(256) void pre_exit_k(const float* __restrict__ pre,
                                                  const float* __restrict__ pim,
                                                  const float* __restrict__ objW,
                                                  float* __restrict__ pe) {
  size_t idx = (size_t)blockIdx.x * 256 + threadIdx.x;   // over NPE
  int hw = (int)(idx & (HW - 1));
  int rest = (int)(idx >> 16);              // b*64 + m*16 + t
  int t = rest & 15, m = (rest >> 4) & 3, b = rest >> 6;
  size_t oi = (((size_t)(b * TT + t)) * HW + hw) * 2;
  float orr = objW[oi], oii = objW[oi + 1];
  float prr = pre[(size_t)m * HW + hw], pii = pim[(size_t)m * HW + hw];
  pe[idx * 2]     = prr * orr - pii * oii;
  pe[idx * 2 + 1] = prr * oii + pii * orr;
}

// ---------------------------------------------------------------------------
// LDS-tiled batched complex GEMM:  Y[img] = A[img?] * B[img?]  (256x256x256).
// A, B, Y complex-interleaved row-major; per-image strides (0 = broadcast
// matrix operand held in L2).  Block tile 64x64, K chunk 16, 8 waves/block,
// 2 16x16 output tiles per wave (A fragment reused), double-buffered LDS.
// Double-buffer selection is done with integer offsets into the single
// __shared__ array (no generic-pointer arrays -> avoids addrspacecast
// static-initializer lowering that ld.lld rejects).
//
// LDS padding chosen for conflict-free fragment reads on 64 banks:
//   A rows:   stride 36 dwords  -> 36*l mod 64 is a 16-lane permutation,
//             and 144B rows keep b128 fragment loads 16B aligned.
//   B k-rows: stride 144 dwords -> lanes 0-15 (K=k) hit even banks 0..30,
//             lanes 16-31 (K=k+2, +288 dwords == 32 mod 64) hit 32..62.
// ---------------------------------------------------------------------------
#define KC    16                 // K chunk
#define NCH   (NN / KC)          // 16 chunks
#define ARS   36                 // A row stride in LDS (dwords)
#define BKS   144                // B k-row stride in LDS (dwords)
#define ABUF  (64 * ARS)         // 2304 dwords
#define BBUF  (KC * BKS)         // 2304 dwords
#define LBUF  (ABUF + BBUF)      // one double-buffer half (dwords)

__global__ __launch_bounds__(256) void gemm_tiled_c(const float* __restrict__ A,
                                                    long aStride,
                                                    const float* __restrict__ B,
                                                    long bStride,
                                                    float* __restrict__ Y) {
  __shared__ float smem[2 * LBUF];               // 36,864 B

  const int tid  = threadIdx.x;
  const int lane = tid & 31;
  const int wid  = tid >> 5;
  const int img  = blockIdx.y;
  const int row0 = (blockIdx.x >> 2) << 6;       // 64-row block
  const int col0 = (blockIdx.x & 3) << 6;        // 64-col block

  const float* Ai = A + (size_t)img * aStride;
  const float* Bi = B + (size_t)img * bStride;

  // Staging assignments (256 threads):
  //   A: thread -> (arow 0..63, aseg 0..3): 4 complex = 2 float4
  //   B: thread -> (bk 0..15, bseg 0..15): 4 complex = 2 float4
  const int arow = tid >> 2, aseg = tid & 3;
  const int bk   = tid >> 4, bseg = tid & 15;
  const float* gA = Ai + ((size_t)(row0 + arow) * NN + aseg * 4) * 2;
  const float* gB = Bi + ((size_t)bk * NN + col0 + bseg * 4) * 2;
  const int lAoff = arow * ARS + aseg * 8;                 // within A sub-buf
  const int lBoff = ABUF + bk * BKS + bseg * 8;            // within half-buf

  // Compute assignment: wave -> row band (wid>>1), column pair (wid&1)
  const int l   = lane & 15;
  const int kh  = (lane >> 4) << 1;              // 0 or 2
  const int r0l = (wid >> 1) << 4;               // local row of wave's tiles
  const int cA  = ((wid & 1) << 5) + l;          // tile 0 column (local)
  const int cB  = cA + 16;                       // tile 1 column (local)

  v8f cre0 = {}, cim0 = {}, cre1 = {}, cim1 = {};

  // ---- stage chunk 0 into half-buffer 0 ----
  {
    float4 a0 = *reinterpret_cast<const float4*>(gA);
    float4 a1 = *reinterpret_cast<const float4*>(gA + 4);
    float4 b0 = *reinterpret_cast<const float4*>(gB);
    float4 b1 = *reinterpret_cast<const float4*>(gB + 4);
    *reinterpret_cast<float4*>(&smem[lAoff])     = a0;
    *reinterpret_cast<float4*>(&smem[lAoff + 4]) = a1;
    *reinterpret_cast<float4*>(&smem[lBoff])     = b0;
    *reinterpret_cast<float4*>(&smem[lBoff + 4]) = b1;
  }
  __syncthreads();

  #pragma unroll 1
  for (int c = 0; c < NCH; ++c) {
    float4 pa0, pa1, pb0, pb1;
    const bool pref = (c + 1 < NCH);
    if (pref) {
      const int kn = (c + 1) * KC;
      pa0 = *reinterpret_cast<const float4*>(gA + (size_t)kn * 2);
      pa1 = *reinterpret_cast<const float4*>(gA + (size_t)kn * 2 + 4);
      pb0 = *reinterpret_cast<const float4*>(gB + (size_t)kn * NN * 2);
      pb1 = *reinterpret_cast<const float4*>(gB + (size_t)kn * NN * 2 + 4);
    }

    const int half = (c & 1) * LBUF;             // current compute half-buf
    #pragma unroll
    for (int kk = 0; kk < KC; kk += 4) {
      // A fragment: rows r0l..r0l+15, K = kk..kk+3 (half-wave split)
      float4 xa = *reinterpret_cast<const float4*>(
          &smem[half + (r0l + l) * ARS + (kk + kh) * 2]);
      v2f are = {xa.x, xa.z};
      v2f aim = {xa.y, xa.w};
      // B fragments for the two N-tiles
      const int bbase = half + ABUF;
      float2 e0a = *reinterpret_cast<const float2*>(&smem[bbase + (kk + kh) * BKS + cA * 2]);
      float2 e1a = *reinterpret_cast<const float2*>(&smem[bbase + (kk + kh + 1) * BKS + cA * 2]);
      float2 e0b = *reinterpret_cast<const float2*>(&smem[bbase + (kk + kh) * BKS + cB * 2]);
      float2 e1b = *reinterpret_cast<const float2*>(&smem[bbase + (kk + kh + 1) * BKS + cB * 2]);
      v2f bre0 = {e0a.x, e1a.x}, bim0 = {e0a.y, e1a.y};
      v2f bre1 = {e0b.x, e1b.x}, bim1 = {e0b.y, e1b.y};
      v2f nbim0 = {-bim0.x, -bim0.y};
      v2f nbim1 = {-bim1.x, -bim1.y};
      cre0 = WMMA_F32(are, bre0, cre0);
      cre0 = WMMA_F32(aim, nbim0, cre0);
      cim0 = WMMA_F32(are, bim0, cim0);
      cim0 = WMMA_F32(aim, bre0, cim0);
      cre1 = WMMA_F32(are, bre1, cre1);
      cre1 = WMMA_F32(aim, nbim1, cre1);
      cim1 = WMMA_F32(are, bim1, cim1);
      cim1 = WMMA_F32(aim, bre1, cim1);
    }

    if (pref) {
      const int nhalf = ((c + 1) & 1) * LBUF;    // next half-buf
      *reinterpret_cast<float4*>(&smem[nhalf + lAoff])     = pa0;
      *reinterpret_cast<float4*>(&smem[nhalf + lAoff + 4]) = pa1;
      *reinterpret_cast<float4*>(&smem[nhalf + lBoff])     = pb0;
      *reinterpret_cast<float4*>(&smem[nhalf + lBoff + 4]) = pb1;
    }
    __syncthreads();
  }

  // ---- store both 16x16 tiles, complex interleaved ----
  const int orow = row0 + r0l + ((lane >> 4) << 3);   // lanes 16-31 -> M+8
  float* Yi = Y + (size_t)img * (HW * 2);
  #pragma unroll
  for (int v = 0; v < 8; ++v) {
    float2 v0 = {cre0[v], cim0[v]};
    float2 v1 = {cre1[v], cim1[v]};
    *reinterpret_cast<float2*>(Yi + ((size_t)(orow + v) * NN + col0 + cA) * 2) = v0;
    *reinterpret_cast<float2*>(Yi + ((size_t)(orow + v) * NN + col0 + cB) * 2) = v1;
  }
}

// intensity = sqrt(sum_m |dif|^2); either scale dif by org/intensity (steps)
// or emit intensity (final pass).
__global__ __launch_bounds__(256) void intensity_k(float* __restrict__ dif,
                                                   const float* __restrict__ org,
                                                   float* __restrict__ inten_out,
                                                   int do_scale) {
  size_t idx = (size_t)blockIdx.x * 256 + threadIdx.x;   // over NOBJ [b,t,h,w]
  int hw = (int)(idx & (HW - 1));
  int rest = (int)(idx >> 16);                // b*16 + t
  int t = rest & 15, b = rest >> 4;
  size_t off[MM];
  float s = 0.0f;
  #pragma unroll
  for (int m = 0; m < MM; ++m) {
    off[m] = (((size_t)((b * MM + m) * TT + t)) * HW + hw) * 2;
    float r = dif[off[m]], i2 = dif[off[m] + 1];
    s += r * r + i2 * i2;
  }
  float inten = sqrtf(s);
  if (do_scale) {
    float ratio = org[idx] / inten;
    #pragma unroll
    for (int m = 0; m < MM; ++m) {
      dif[off[m]]     *= ratio;
      dif[off[m] + 1] *= ratio;
    }
  } else {
    inten_out[idx] = inten;
  }
}

// objects += alpha * combine(|u|*mask, angle(u)*mask), with
// u = sum_m conj(probe)*(exitw - probe*objects) / prob_abs.
// pre_exit is recomputed here (objects not yet updated) to save a buffer.
__global__ __launch_bounds__(256) void update_k(const float* __restrict__ pre,
                                                const float* __restrict__ pim,
                                                const float* __restrict__ exitw,
                                                const float* __restrict__ mask,
                                                const float* __restrict__ alpha,
                                                const float* __restrict__ pa,
                                                float* __restrict__ objW) {
  size_t idx = (size_t)blockIdx.x * 256 + threadIdx.x;   // over NOBJ
  int hw = (int)(idx & (HW - 1));
  int rest = (int)(idx >> 16);
  int t = rest & 15, b = rest >> 4;
  size_t oi = idx * 2;
  float orr = objW[oi], oii = objW[oi + 1];
  float ur = 0.0f, ui = 0.0f;
  #pragma unroll
  for (int m = 0; m < MM; ++m) {
    float prr = pre[(size_t)m * HW + hw], pii = pim[(size_t)m * HW + hw];
    float per = prr * orr - pii * oii;          // recomputed pre_exit
    float pei = prr * oii + pii * orr;
    size_t off = (((size_t)((b * MM + m) * TT + t)) * HW + hw) * 2;
    float dr = exitw[off] - per;
    float di = exitw[off + 1] - pei;
    ur += prr * dr + pii * di;                  // conj(p) * dexit
    ui += prr * di - pii * dr;
  }
  float inv = 1.0f / pa[0];
  ur *= inv; ui *= inv;
  float mv = mask[hw];
  float amp = sqrtf(ur * ur + ui * ui) * mv;
  float ph  = atan2f(ui, ur) * mv;
  float cs, sn;
  __sincosf(ph, &sn, &cs);
  float a = alpha[0];
  objW[oi]     = orr + a * amp * cs;
  objW[oi + 1] = oii + a * amp * sn;
}

// ---------------------------------------------------------------------------
extern "C" void kernel_launch(void* const* d_in, const int* in_sizes, int n_in,
                              void* d_out, int out_size, void* d_ws, size_t ws_size,
                              hipStream_t stream) {
  const float* objects_re = (const float*)d_in[0];
  const float* objects_im = (const float*)d_in[1];
  const float* probe_re   = (const float*)d_in[2];
  const float* probe_im   = (const float*)d_in[3];
  const float* org        = (const float*)d_in[4];
  const float* mask       = (const float*)d_in[5];
  const float* alpha      = (const float*)d_in[6];
  (void)in_sizes; (void)n_in; (void)out_size; (void)ws_size;

  // Workspace layout (floats)
  float* ws = (float*)d_ws;
  const size_t MAT2 = (size_t)NN * NN * 2;     // one interleaved complex matrix
  float* FsC  = ws + 0 * MAT2;
  float* FsTC = ws + 1 * MAT2;
  float* GsC  = ws + 2 * MAT2;
  float* GsTC = ws + 3 * MAT2;
  float* pa   = ws + 4 * MAT2;                 // 1 scalar (+15 pad)
  float* objW = pa + 16;                       // 2*NOBJ floats, interleaved
  float* bufA = objW + 2 * NOBJ;               // 2*NPE floats
  float* bufB = bufA + 2 * NPE;                // 2*NPE floats

  const dim3 gemmGrid(16, NIMG);               // 4x4 block tiles of 64x64
  const int  blk = 256;
  const int  gPE  = (int)(NPE  / blk);         // 65536 blocks
  const int  gOBJ = (int)(NOBJ / blk);         // 16384 blocks
  const long IMS  = (long)HW * 2;              // per-image stride (floats)

  init_dft_k<<<dim3(NN), NN, 0, stream>>>(FsC, FsTC, GsC, GsTC);
  prob_abs_k<<<1, blk, 0, stream>>>(probe_re, probe_im, pa);
  copy_obj_k<<<gOBJ, blk, 0, stream>>>(objects_re, objects_im, objW);

  for (int step = 0; step < 6; ++step) {
    // pre_exit -> bufB
    pre_exit_k<<<gPE, blk, 0, stream>>>(probe_re, probe_im, objW, bufB);
    // dif = Fs * pre_exit * Fs^T   (== fftshift(fft2)/fc)
    gemm_tiled_c<<<gemmGrid, blk, 0, stream>>>(bufB, IMS, FsTC, 0, bufA);
    gemm_tiled_c<<<gemmGrid, blk, 0, stream>>>(FsC, 0, bufA, IMS, bufB);
    if (step < 5) {
      intensity_k<<<gOBJ, blk, 0, stream>>>(bufB, org, nullptr, 1);
      // exitw = Gs * dif * Gs^T   (== ifft2(ifftshift)*fc)
      gemm_tiled_c<<<gemmGrid, blk, 0, stream>>>(bufB, IMS, GsTC, 0, bufA);
      gemm_tiled_c<<<gemmGrid, blk, 0, stream>>>(GsC, 0, bufA, IMS, bufB);
      update_k<<<gOBJ, blk, 0, stream>>>(probe_re, probe_im, bufB, mask, alpha, pa, objW);
    } else {
      // final: intensity -> d_out[0:NOBJ], objects -> d_out[NOBJ:] interleaved
      intensity_k<<<gOBJ, blk, 0, stream>>>(bufB, org, (float*)d_out, 0);
      (void)hipMemcpyAsync((float*)d_out + NOBJ, objW, 2 * NOBJ * sizeof(float),
                           hipMemcpyDeviceToDevice, stream);
    }
  }
}